// CausalSelfAttention_34866544509539
// MI455X (gfx1250) — compile-verified
//
#include <hip/hip_runtime.h>
#include <math.h>

// ---------------------------------------------------------------------------
// CDNA5 (gfx1250) f32 WMMA causal self-attention.
//   DIM=2048, HEADS=16, head_dim=128, BATCH=2, SEQ=2048  ->  M=B*S=4096 rows.
// All matrix math runs on V_WMMA_F32_16X16X4_F32 (full f32 matrix pipe).
// ---------------------------------------------------------------------------

#define SEQ_C   2048
#define DIM_C   2048
#define HEADS_C 16
#define HD_C    128            // head dim
#define MROWS   4096           // BATCH * SEQ

typedef float v2f __attribute__((ext_vector_type(2)));
typedef float v8f __attribute__((ext_vector_type(8)));

__device__ __forceinline__ v8f wmma_f32(v2f a, v2f b, v8f c) {
  // D = A(16x4) * B(4x16) + C(16x16), all f32.
  return __builtin_amdgcn_wmma_f32_16x16x4_f32(
      /*neg_a=*/false, a, /*neg_b=*/false, b,
      /*c_mod=*/(short)0, c, /*reuse_a=*/false, /*reuse_b=*/false);
}

__device__ __forceinline__ v8f v8f_zero() {
  v8f z;
#pragma unroll
  for (int i = 0; i < 8; ++i) z[i] = 0.0f;
  return z;
}

// ---------------------------------------------------------------------------
// GEMM: C[M,N] = A[M,K] @ B[K,N], row-major f32, via WMMA 16x16x4.
// Block = 256 threads = 8 waves. Wave (wm,wn) computes a 16x64 strip:
//   4 accumulator tiles sharing one A fragment per k-step (A reuse).
// Block tile = 64(M) x 128(N). Grid = (M/64, N/128).
// ---------------------------------------------------------------------------
__global__ __launch_bounds__(256) void gemm_f32_wmma(
    const float* __restrict__ A, const float* __restrict__ B,
    float* __restrict__ C, int M, int N, int K) {
  const int lane = threadIdx.x & 31;
  const int wave = threadIdx.x >> 5;
  const int lid  = lane & 15;      // M/N index within tile
  const int lg   = lane >> 4;      // lanegroup: selects K pair / row half
  const int wm   = wave & 3;
  const int wn   = wave >> 2;
  const int m0   = blockIdx.x * 64  + wm * 16;
  const int n0   = blockIdx.y * 128 + wn * 64;

  v8f acc[4];
#pragma unroll
  for (int t = 0; t < 4; ++t) acc[t] = v8f_zero();

  const float* arow = A + (size_t)(m0 + lid) * K;

  for (int k = 0; k < K; k += 8) {
    // A fragments: A[m0+lid][k + 2*lg + {0,1}] (+4 for second k-step)
    v2f a0 = *(const v2f*)(arow + k + 2 * lg);
    v2f a1 = *(const v2f*)(arow + k + 4 + 2 * lg);
#pragma unroll
    for (int t = 0; t < 4; ++t) {
      const float* bcol = B + (size_t)(k + 2 * lg) * N + n0 + t * 16 + lid;
      v2f b0, b1;
      b0.x = bcol[0];
      b0.y = bcol[N];
      b1.x = bcol[4 * (size_t)N];
      b1.y = bcol[5 * (size_t)N];
      acc[t] = wmma_f32(a0, b0, acc[t]);
      acc[t] = wmma_f32(a1, b1, acc[t]);
    }
  }

  // D layout: vgpr i, lanegroup lg -> row = i + 8*lg, col = lid.
#pragma unroll
  for (int t = 0; t < 4; ++t)
#pragma unroll
    for (int i = 0; i < 8; ++i)
      C[(size_t)(m0 + i + 8 * lg) * N + n0 + t * 16 + lid] = acc[t][i];
}

// ---------------------------------------------------------------------------
// Attention: per (batch,head), flash-style streaming softmax.
// One wave owns a 16-row query block, iterates 16-key blocks.
// Mask is faithful to the reference: KEEP col > row (strictly upper),
// masked value = -FLT_MAX (so fully-masked rows -> uniform, like the ref).
// ---------------------------------------------------------------------------
__global__ __launch_bounds__(256, 1) void attn_f32_wmma(
    const float* __restrict__ Q, const float* __restrict__ Km,
    const float* __restrict__ V, float* __restrict__ O) {
  __shared__ __align__(16) float Pl[8][16 * 18];  // per-wave 16x16 tile, pad 18

  const int lane = threadIdx.x & 31;
  const int wave = threadIdx.x >> 5;
  const int lid  = lane & 15;
  const int lg   = lane >> 4;

  const int bh = blockIdx.x;           // 0..31
  const int b  = bh >> 4;
  const int h  = bh & 15;
  const int qb = blockIdx.y * 8 + wave;  // 0..127
  const int q0 = qb * 16;
  const int d0 = h * HD_C;
  const size_t rowbase = (size_t)b * SEQ_C;

  // Preload Q block as A-fragments: qa[s] covers K-dim = 4*s + 2*lg + {0,1}.
  const float* qrow = Q + (rowbase + q0 + lid) * DIM_C + d0;
  v2f qa[32];
#pragma unroll
  for (int s = 0; s < 32; ++s) qa[s] = *(const v2f*)(qrow + 4 * s + 2 * lg);

  v8f o[8];
#pragma unroll
  for (int dt = 0; dt < 8; ++dt) o[dt] = v8f_zero();
  float m[8], l[8];
#pragma unroll
  for (int i = 0; i < 8; ++i) { m[i] = -INFINITY; l[i] = 0.0f; }

  const float scale = 0.08838834764831845f;  // 1/sqrt(128)
  const float MASKV = -3.402823466e38f;      // -FLT_MAX (matches reference)

  for (int kb = 0; kb < SEQ_C / 16; ++kb) {
    const int kk0 = kb * 16;

    // ---- S = Q @ K^T : B-frag b.x = K[kk0+lid][dd + 2*lg] ----
    v8f s = v8f_zero();
    const float* krow = Km + (rowbase + kk0 + lid) * DIM_C + d0;
#pragma unroll
    for (int ss = 0; ss < 32; ++ss) {
      v2f bf = *(const v2f*)(krow + 4 * ss + 2 * lg);
      s = wmma_f32(qa[ss], bf, s);
    }

    // ---- mask + online softmax update ----
    float alpha[8];
#pragma unroll
    for (int i = 0; i < 8; ++i) {
      const int row = q0 + i + 8 * lg;
      const int col = kk0 + lid;
      float sv = s[i] * scale;
      if (!(col > row)) sv = MASKV;  // keep strictly-upper only
      // row max across the 16 lanes holding this row
      float t = sv;
      t = fmaxf(t, __shfl_xor(t, 1, 16));
      t = fmaxf(t, __shfl_xor(t, 2, 16));
      t = fmaxf(t, __shfl_xor(t, 4, 16));
      t = fmaxf(t, __shfl_xor(t, 8, 16));
      const float mnew = fmaxf(m[i], t);
      alpha[i] = expf(m[i] - mnew);
      const float pv = expf(sv - mnew);
      m[i] = mnew;
      float rs = pv;
      rs += __shfl_xor(rs, 1, 16);
      rs += __shfl_xor(rs, 2, 16);
      rs += __shfl_xor(rs, 4, 16);
      rs += __shfl_xor(rs, 8, 16);
      l[i] = l[i] * alpha[i] + rs;
      Pl[wave][(i + 8 * lg) * 18 + lid] = pv;  // stash P tile (D layout)
    }

    // rescale running output
#pragma unroll
    for (int dt = 0; dt < 8; ++dt)
#pragma unroll
      for (int i = 0; i < 8; ++i) o[dt][i] *= alpha[i];

    // ---- reload P as A-fragments (D layout -> A layout via LDS) ----
    v2f pa[4];
#pragma unroll
    for (int kk = 0; kk < 4; ++kk)
      pa[kk] = *(const v2f*)&Pl[wave][lid * 18 + kk * 4 + 2 * lg];

    // ---- O += P @ V ----
#pragma unroll
    for (int dt = 0; dt < 8; ++dt) {
      const float* vcol = V + (rowbase + kk0) * DIM_C + d0 + dt * 16 + lid;
#pragma unroll
      for (int kk = 0; kk < 4; ++kk) {
        v2f bf;
        bf.x = vcol[(size_t)(kk * 4 + 2 * lg) * DIM_C];
        bf.y = vcol[(size_t)(kk * 4 + 2 * lg + 1) * DIM_C];
        o[dt] = wmma_f32(pa[kk], bf, o[dt]);
      }
    }
  }

  // ---- normalize and store (b, s, h*d) layout: un-transposed directly ----
#pragma unroll
  for (int i = 0; i < 8; ++i) {
    const float inv = 1.0f / l[i];
    float* orow = O + (rowbase + q0 + i + 8 * lg) * DIM_C + d0;
#pragma unroll
    for (int dt = 0; dt < 8; ++dt) orow[dt * 16 + lid] = o[dt][i] * inv;
  }
}

// ---------------------------------------------------------------------------
// Launch: Q/K/V projections -> attention -> output projection.
// Workspace: Q | K | V | attn_out, 32 MB each (128 MB total).
// ---------------------------------------------------------------------------
extern "C" void kernel_launch(void* const* d_in, const int* in_sizes, int n_in,
                              void* d_out, int out_size, void* d_ws, size_t ws_size,
                              hipStream_t stream) {
  const float* x  = (const float*)d_in[0];
  const float* Wq = (const float*)d_in[1];
  const float* Wk = (const float*)d_in[2];
  const float* Wv = (const float*)d_in[3];
  const float* Wo = (const float*)d_in[4];
  float* out = (float*)d_out;

  const size_t mat = (size_t)MROWS * DIM_C;  // 8M floats = 32MB
  float* Qb = (float*)d_ws;
  float* Kb = Qb + mat;
  float* Vb = Kb + mat;
  float* Ab = Vb + mat;

  dim3 gblk(256);
  dim3 ggrid(MROWS / 64, DIM_C / 128);
  gemm_f32_wmma<<<ggrid, gblk, 0, stream>>>(x, Wq, Qb, MROWS, DIM_C, DIM_C);
  gemm_f32_wmma<<<ggrid, gblk, 0, stream>>>(x, Wk, Kb, MROWS, DIM_C, DIM_C);
  gemm_f32_wmma<<<ggrid, gblk, 0, stream>>>(x, Wv, Vb, MROWS, DIM_C, DIM_C);

  dim3 agrid(2 * HEADS_C, SEQ_C / (16 * 8));
  attn_f32_wmma<<<agrid, gblk, 0, stream>>>(Qb, Kb, Vb, Ab);

  gemm_f32_wmma<<<ggrid, gblk, 0, stream>>>(Ab, Wo, out, MROWS, DIM_C, DIM_C);
}